// DAN_81690277970359
// MI455X (gfx1250) — compile-verified
//
#include <hip/hip_runtime.h>
#include <math.h>

typedef float v2f __attribute__((ext_vector_type(2)));
typedef float v8f __attribute__((ext_vector_type(8)));

#define T_LEN 36
#define NP    80
#define P_DIM 161        // 2N+1
#define BATCH 256
#define DDIM  50
#define NCOL  12800      // BATCH*DDIM
#define KPAD  164        // P_DIM padded to mult of 4 (WMMA K step)
#define MPAD  176        // P_DIM padded to mult of 16 (WMMA M tiles)
#define MAX_IT 100
#define TOLV  1e-5f
#define LAMV  0.1f

// y buffers use a row-pair interleaved layout so the WMMA B-fragment
// (K-adjacent pair per lane) is one aligned b64 load:
//   y[k][n]  ->  buf[(k>>1)*(2*NCOL) + n*2 + (k&1)]
// Rows 0..163 -> 82 pairs; total size unchanged (164*12800 floats).

// ---- workspace layout (float units) ----
#define OFF_APAD 0ul                 // 176*164      = 28864
#define OFF_X    28864ul             // 161*12800    = 2060800
#define OFF_Y0   2089664ul           // 164*12800    = 2099200
#define OFF_Y1   4188864ul           // 164*12800    = 2099200
#define OFF_B    6288064ul           // 161*12800    = 2060800
#define OFF_D    8348864ul           // 36*161       = 5796
#define OFF_DTD  8354660ul           // 161*161      = 25921
#define OFF_SCAL 8380581ul           // [0]=1/L, [1]=thr
#define OFF_PART 8380583ul           // 1100 block partials
#define OFF_DONE 8381683ul           // done flag (int)
#define WS_FLOATS 8381684ul          // ~33.5 MB

// ---------- zero-init mutable state each launch ----------
__global__ void k_init(float* ws) {
    size_t i = (size_t)blockIdx.x * 256 + threadIdx.x;
    const size_t NZ = OFF_B - OFF_X;          // x + y0 + y1 = 6259200
    if (i < NZ) {
        ws[OFF_X + i] = 0.0f;
    } else {
        size_t j = i - NZ;
        if (j < 1101) ws[OFF_PART + j] = 0.0f; // partials + done flag
    }
}

// ---------- dictionary: (36 x 161), column-normalized except col 0 ----------
__global__ void k_dic(const float* __restrict__ rho, const float* __restrict__ theta,
                      float* __restrict__ D, float* __restrict__ dic_out) {
    int j = threadIdx.x;
    if (j >= P_DIM) return;
    float vals[T_LEN];
    if (j == 0) {
        for (int t = 0; t < T_LEN; t++) vals[t] = 1.0f;
    } else {
        int pole   = (j <= NP) ? (j - 1) : (j - 1 - NP);
        bool is_cos = (j <= NP);
        float r  = 0.001f + 1.149f / (1.0f + expf(-rho[pole]));
        float th = 3.14159265358979323846f / (1.0f + expf(-theta[pole]));
        float p = 1.0f, ss = 0.0f;
        for (int t = 0; t < T_LEN; t++) {
            float ang = th * (float)t;
            float v = p * (is_cos ? cosf(ang) : sinf(ang));
            vals[t] = v; ss += v * v; p *= r;
        }
        float nrm = sqrtf(ss);
        for (int t = 0; t < T_LEN; t++) vals[t] = vals[t] / nrm;
    }
    for (int t = 0; t < T_LEN; t++) {
        D[t * P_DIM + j] = vals[t];
        dic_out[t * P_DIM + j] = vals[t];
    }
}

// ---------- DtD = D^T D (161 x 161) ----------
__global__ void k_dtd(const float* __restrict__ D, float* __restrict__ DtD) {
    int i = blockIdx.x * 256 + threadIdx.x;
    if (i >= P_DIM * P_DIM) return;
    int p = i / P_DIM, q = i % P_DIM;
    float s = 0.0f;
    for (int t = 0; t < T_LEN; t++) s += D[t * P_DIM + p] * D[t * P_DIM + q];
    DtD[i] = s;
}

// ---------- spectral norm of DtD via power iteration (deterministic) ----------
__global__ void k_power(const float* __restrict__ DtD, float* __restrict__ scal) {
    __shared__ float v[P_DIM];
    __shared__ float red[256];
    int tid = threadIdx.x;
    if (tid < P_DIM) v[tid] = 0.07881104f;   // 1/sqrt(161)
    __syncthreads();
    float lam = 1.0f;
    for (int it = 0; it < 300; it++) {
        float w = 0.0f;
        if (tid < P_DIM) {
            const float* row = DtD + tid * P_DIM;
            for (int q = 0; q < P_DIM; q++) w += row[q] * v[q];
        }
        red[tid] = (tid < P_DIM) ? w * w : 0.0f;
        __syncthreads();
        for (int s = 128; s > 0; s >>= 1) { if (tid < s) red[tid] += red[tid + s]; __syncthreads(); }
        lam = sqrtf(red[0]);
        __syncthreads();
        if (tid < P_DIM) v[tid] = w / lam;
        __syncthreads();
    }
    if (tid == 0) { scal[0] = 1.0f / lam; scal[1] = LAMV / lam; }
}

// ---------- Apad = I - DtD/L, zero-padded to 176x164 ----------
__global__ void k_amat(const float* __restrict__ DtD, const float* __restrict__ scal,
                       float* __restrict__ Apad) {
    int i = blockIdx.x * 256 + threadIdx.x;
    if (i >= MPAD * KPAD) return;
    int m = i / KPAD, k = i % KPAD;
    float v = 0.0f;
    if (m < P_DIM && k < P_DIM)
        v = ((m == k) ? 1.0f : 0.0f) - scal[0] * DtD[m * P_DIM + k];
    Apad[i] = v;
}

// ---------- Bm[p][n] = (1/L) * sum_t D[t][p] * x[b][t][dd],  n=b*50+dd ----------
__global__ void k_bmat(const float* __restrict__ D, const float* __restrict__ xin,
                       const float* __restrict__ scal, float* __restrict__ Bm) {
    int i = blockIdx.x * 256 + threadIdx.x;   // < 161*12800
    int p = i / NCOL, n = i % NCOL;
    int b = n / DDIM, dd = n % DDIM;
    const float* xb = xin + (size_t)b * T_LEN * DDIM + dd;
    float s = 0.0f;
    for (int t = 0; t < T_LEN; t++) s += D[t * P_DIM + p] * xb[t * DDIM];
    Bm[i] = scal[0] * s;
}

// ---------- one FISTA iteration: x_new = shrink(A@y + B); y momentum; diff partials ----
__global__ void __launch_bounds__(256)
k_fista(const float* __restrict__ A, const float* __restrict__ Bm,
        float* __restrict__ x, const float* __restrict__ yin,
        float* __restrict__ yout, const float* __restrict__ scal,
        const int* __restrict__ done, float* __restrict__ partial, float tt) {
    int tid = threadIdx.x, bid = blockIdx.x;
    int wave = tid >> 5, lane = tid & 31;
    int mtile = bid % 11;
    int ntile = (bid / 11) * 8 + wave;
    int lan16 = lane & 15;
    int ncol  = ntile * 16 + lan16;
    float dsq = 0.0f;
    int dn = *done;
    if (!dn) {
        v8f acc = {0.f, 0.f, 0.f, 0.f, 0.f, 0.f, 0.f, 0.f};
        int klo = (lane < 16) ? 0 : 2;
        // A frag: lane L<16 -> row M=L, K = k0+{0,1};  L>=16 -> row M=L-16, K = k0+{2,3}
        const float* Arow = A + (size_t)(mtile * 16 + lan16) * KPAD + klo;
        // B frag: lane holds column N=lane%16; K pair (k0+klo, k0+klo+1) is one b64
        // in the row-pair interleaved y layout (k0+klo is always even).
        const float* ypair = yin + (size_t)ncol * 2;
        for (int k0 = 0; k0 < P_DIM; k0 += 4) {
            v2f a = *(const v2f*)(Arow + k0);
            v2f b = *(const v2f*)(ypair + (size_t)((k0 + klo) >> 1) * (2 * NCOL));
            acc = __builtin_amdgcn_wmma_f32_16x16x4_f32(
                false, a, false, b, (short)0, acc, false, false);
        }
        float thr = scal[1];
        int mbase = mtile * 16 + ((lane < 16) ? 0 : 8);
        #pragma unroll
        for (int i = 0; i < 8; i++) {
            int m = mbase + i;
            if (m < P_DIM) {
                size_t off = (size_t)m * NCOL + ncol;
                float z  = acc[i] + Bm[off];
                float az = fabsf(z) - thr;
                float xn = (az > 0.0f) ? copysignf(az, z) : 0.0f;
                float xo = x[off];
                float df = xo - xn;
                dsq += df * df;
                x[off] = xn;
                size_t yoff = (size_t)(m >> 1) * (2 * NCOL) + (size_t)ncol * 2 + (m & 1);
                yout[yoff] = (1.0f + tt) * xn - tt * xo;
            }
        }
    } else {
        // frozen: propagate y through the ping-pong (same packed slots)
        int mbase = mtile * 16 + ((lane < 16) ? 0 : 8);
        #pragma unroll
        for (int i = 0; i < 8; i++) {
            int m = mbase + i;
            if (m < P_DIM) {
                size_t yoff = (size_t)(m >> 1) * (2 * NCOL) + (size_t)ncol * 2 + (m & 1);
                yout[yoff] = yin[yoff];
            }
        }
    }
    __shared__ float red[256];
    red[tid] = dsq;
    __syncthreads();
    for (int s = 128; s > 0; s >>= 1) { if (tid < s) red[tid] += red[tid + s]; __syncthreads(); }
    if (tid == 0) partial[bid] = red[0];
}

// ---------- finalize diff, update done flag ----------
__global__ void k_done(const float* __restrict__ partial, int* __restrict__ done) {
    __shared__ float red[256];
    int tid = threadIdx.x;
    float s = 0.0f;
    for (int j = tid; j < 1100; j += 256) s += partial[j];
    red[tid] = s;
    __syncthreads();
    for (int st = 128; st > 0; st >>= 1) { if (tid < st) red[tid] += red[tid + st]; __syncthreads(); }
    if (tid == 0) {
        float diff = sqrtf(red[0]) / (float)DDIM;
        if (diff < TOLV) *done = 1;
    }
}

// ---------- C output: (B, P, d) from x[p][b*50+dd] ----------
__global__ void k_copyc(const float* __restrict__ x, float* __restrict__ out) {
    int i = blockIdx.x * 256 + threadIdx.x;   // < 256*161*50
    int b = i / (P_DIM * DDIM);
    int rem = i % (P_DIM * DDIM);
    int p = rem / DDIM, dd = rem % DDIM;
    out[i] = x[(size_t)p * NCOL + b * DDIM + dd];
}

// ---------- reconst[b][t][dd] = sum_p D[t][p] * C[b][p][dd] ----------
__global__ void k_reconst(const float* __restrict__ D, const float* __restrict__ x,
                          float* __restrict__ out) {
    int i = blockIdx.x * 256 + threadIdx.x;   // < 256*36*50
    int b = i / (T_LEN * DDIM);
    int rem = i % (T_LEN * DDIM);
    int t = rem / DDIM, dd = rem % DDIM;
    const float* xc = x + b * DDIM + dd;
    float s = 0.0f;
    for (int p = 0; p < P_DIM; p++) s += D[t * P_DIM + p] * xc[(size_t)p * NCOL];
    out[i] = s;
}

extern "C" void kernel_launch(void* const* d_in, const int* in_sizes, int n_in,
                              void* d_out, int out_size, void* d_ws, size_t ws_size,
                              hipStream_t stream) {
    const float* x_in  = (const float*)d_in[0];   // (256, 36, 50)
    const float* rho   = (const float*)d_in[1];   // (80,)
    const float* theta = (const float*)d_in[2];   // (80,)
    float* out = (float*)d_out;                   // C | dic | reconst
    float* ws  = (float*)d_ws;
    if (ws_size < WS_FLOATS * sizeof(float)) return;

    float* Apad = ws + OFF_APAD;
    float* X    = ws + OFF_X;
    float* Y0   = ws + OFF_Y0;
    float* Y1   = ws + OFF_Y1;
    float* Bm   = ws + OFF_B;
    float* D    = ws + OFF_D;
    float* DtD  = ws + OFF_DTD;
    float* scal = ws + OFF_SCAL;
    float* part = ws + OFF_PART;
    int*   done = (int*)(ws + OFF_DONE);

    const size_t initN = (OFF_B - OFF_X) + 1101;
    k_init<<<(unsigned)((initN + 255) / 256), 256, 0, stream>>>(ws);
    k_dic<<<1, 192, 0, stream>>>(rho, theta, D, out + 2060800);
    k_dtd<<<(P_DIM * P_DIM + 255) / 256, 256, 0, stream>>>(D, DtD);
    k_power<<<1, 256, 0, stream>>>(DtD, scal);
    k_amat<<<(MPAD * KPAD + 255) / 256, 256, 0, stream>>>(DtD, scal, Apad);
    k_bmat<<<(P_DIM * NCOL) / 256, 256, 0, stream>>>(D, x_in, scal, Bm);

    // t / momentum sequence is data-independent: precompute on host (fp32 like ref)
    float t = 1.0f;
    float* yin = Y0; float* yout = Y1;
    for (int k = 0; k < MAX_IT; k++) {
        float tn = (1.0f + sqrtf(1.0f + 4.0f * t * t)) * 0.5f;
        float tt = (t - 1.0f) / tn;
        k_fista<<<1100, 256, 0, stream>>>(Apad, Bm, X, yin, yout, scal, done, part, tt);
        k_done<<<1, 256, 0, stream>>>(part, done);
        float* tmp = yin; yin = yout; yout = tmp;
        t = tn;
    }

    k_copyc<<<(P_DIM * NCOL) / 256, 256, 0, stream>>>(X, out);
    k_reconst<<<(BATCH * T_LEN * DDIM) / 256, 256, 0, stream>>>(D, X, out + 2066596);
}